// Method_GCN_12678743458316
// MI455X (gfx1250) — compile-verified
//
#include <hip/hip_runtime.h>
#include <hip/hip_bf16.h>

// ---------------------------------------------------------------------------
// 2-layer GCN for MI455X (gfx1250, wave32).
//   agg = S(h*norm by dst) + S(h*norm by src) + h/deg ; h' = relu(agg @ W^T)
// Strategy: L2-resident gather/scatter (one wave per edge, float atomics),
// fp32 WMMA (v_wmma_f32_16x16x4_f32) for the dense transforms.
// ---------------------------------------------------------------------------

typedef float v2f __attribute__((ext_vector_type(2)));
typedef float v8f __attribute__((ext_vector_type(8)));

static __device__ __forceinline__ v2f ldv2(const float* p) {
    return *reinterpret_cast<const v2f*>(p);   // 8B-aligned (even fp32 offsets)
}

// ---------------- elementwise helpers ----------------

__global__ void k_zero_f32(float* __restrict__ p, int n) {
    int i = blockIdx.x * blockDim.x + threadIdx.x;
    if (i < n) p[i] = 0.0f;
}

__global__ void k_clamp1_f32(float* __restrict__ p, int n) {
    int i = blockIdx.x * blockDim.x + threadIdx.x;
    if (i < n) p[i] = fmaxf(p[i], 1.0f);
}

// deg[src]+=1, deg[dst]+=1 per edge
__global__ void k_degree(const int* __restrict__ edges, float* __restrict__ deg, int nEdges) {
    int e = blockIdx.x * blockDim.x + threadIdx.x;
    if (e < nEdges) {
        atomicAdd(&deg[edges[2 * e + 0]], 1.0f);
        atomicAdd(&deg[edges[2 * e + 1]], 1.0f);
    }
}

// agg[i][f] = h[i][f] / deg[i]   (the "+ h/deg" self term, folded into init)
template <int D>
__global__ void k_init_agg(const float* __restrict__ h, const float* __restrict__ deg,
                           float* __restrict__ agg, int nNodes) {
    long long i = (long long)blockIdx.x * blockDim.x + threadIdx.x;
    long long total = (long long)nNodes * D;
    if (i < total) {
        int node = (int)(i / D);
        agg[i] = h[i] / deg[node];
    }
}

// ---------------- edge aggregation: one wave32 per edge ----------------
// norm = rsqrt(deg[s]*deg[d]); agg[d] += h[s]*norm; agg[s] += h[d]*norm
// Each lane covers VEC = D/32 contiguous features (coalesced 512B/256B rows).
template <int D>
__global__ void k_edge_agg(const float* __restrict__ h, const int* __restrict__ edges,
                           const float* __restrict__ deg, float* __restrict__ agg,
                           int nEdges) {
    constexpr int VEC = D / 32;
    int wave = (blockIdx.x * blockDim.x + threadIdx.x) >> 5;
    if (wave >= nEdges) return;                    // wave-uniform guard
    int lane = threadIdx.x & 31;

    int s = edges[2 * wave + 0];
    int d = edges[2 * wave + 1];
    float nrm = rsqrtf(deg[s] * deg[d]);

    int f0 = lane * VEC;
    const float* hs = h + (size_t)s * D + f0;
    const float* hd = h + (size_t)d * D + f0;
    float* as = agg + (size_t)s * D + f0;
    float* ad = agg + (size_t)d * D + f0;

#pragma unroll
    for (int v = 0; v < VEC; ++v) {
        atomicAdd(ad + v, hs[v] * nrm);
        atomicAdd(as + v, hd[v] * nrm);
    }
}

// ---------------- WMMA fp32 GEMM + ReLU ----------------
// O(M x N) = relu(A(M x K) @ W(N x K)^T); one wave per 16x16 output tile.
// V_WMMA_F32_16X16X4_F32 fragment mapping (ISA 7.12.2):
//   A: lane L -> A[m0 + (L&15)][k + 2*(L>>4) + {0,1}]   (float2)
//   B: lane L -> B[k + 2*(L>>4) + {0,1}][n0 + (L&15)] = W[n0+(L&15)][k+2*(L>>4)+{0,1}]
//   C/D: vgpr r, lane L -> O[m0 + r + 8*(L>>4)][n0 + (L&15)]
template <int K, int N>
__global__ void k_gemm_relu_wmma(const float* __restrict__ A, const float* __restrict__ W,
                                 float* __restrict__ O, int Mtiles) {
    constexpr int NT = N / 16;
    int wave = (blockIdx.x * blockDim.x + threadIdx.x) >> 5;
    int mt = wave / NT;
    int nt = wave % NT;
    if (mt >= Mtiles) return;                      // wave-uniform: EXEC stays all-1s

    int lane = threadIdx.x & 31;
    int half = lane >> 4;                          // 0: K+{0,1}; 1: K+{2,3}
    int l15  = lane & 15;

    const float* arow = A + (size_t)(mt * 16 + l15) * K;
    const float* wrow = W + (size_t)(nt * 16 + l15) * K;

    v8f c = {};
#pragma unroll
    for (int k = 0; k < K; k += 4) {
        int ka = k + 2 * half;
        v2f a = ldv2(arow + ka);
        v2f b = ldv2(wrow + ka);
        // 8 args: (neg_a, A, neg_b, B, c_mod, C, reuse_a, reuse_b)
        c = __builtin_amdgcn_wmma_f32_16x16x4_f32(false, a, false, b, (short)0, c,
                                                  false, false);
    }

    float* orow = O + (size_t)(mt * 16 + 8 * half) * N + nt * 16 + l15;
#pragma unroll
    for (int r = 0; r < 8; ++r) {
        float v = c[r];
        orow[(size_t)r * N] = v > 0.0f ? v : 0.0f;
    }
}

// ---------------------------------------------------------------------------

extern "C" void kernel_launch(void* const* d_in, const int* in_sizes, int n_in,
                              void* d_out, int out_size, void* d_ws, size_t ws_size,
                              hipStream_t stream) {
    const float* x     = (const float*)d_in[0];   // (nNodes, 128)
    const int*   edges = (const int*)d_in[1];     // (nEdges, 2) int32
    const float* W1    = (const float*)d_in[2];   // (64, 128)
    const float* W2    = (const float*)d_in[3];   // (16, 64)
    float*       out   = (float*)d_out;           // (nNodes, 16)

    const int nNodes = in_sizes[0] / 128;         // 50000 (divisible by 16)
    const int nEdges = in_sizes[1] / 2;           // 800000

    // workspace layout: deg | agg (nNodes*128, reused as agg2) | h1 (nNodes*64)
    char* ws = (char*)d_ws;
    float* deg = (float*)ws;
    size_t off = ((size_t)nNodes * sizeof(float) + 255) & ~(size_t)255;
    float* agg1 = (float*)(ws + off);
    off += (size_t)nNodes * 128 * sizeof(float);
    float* h1 = (float*)(ws + off);
    float* agg2 = agg1;                           // layer-1 agg dead after GEMM

    const int B = 256;                            // 8 waves / block
    auto cdiv = [](long long a, long long b) { return (int)((a + b - 1) / b); };

    // ---- degrees ----
    k_zero_f32<<<cdiv(nNodes, B), B, 0, stream>>>(deg, nNodes);
    k_degree<<<cdiv(nEdges, B), B, 0, stream>>>(edges, deg, nEdges);
    k_clamp1_f32<<<cdiv(nNodes, B), B, 0, stream>>>(deg, nNodes);

    // ---- layer 1: D=128 -> 64 ----
    k_init_agg<128><<<cdiv((long long)nNodes * 128, B), B, 0, stream>>>(x, deg, agg1, nNodes);
    k_edge_agg<128><<<cdiv(nEdges, 8), B, 0, stream>>>(x, edges, deg, agg1, nEdges);
    {
        int mtiles = nNodes / 16;                 // 3125
        int waves = mtiles * (64 / 16);
        k_gemm_relu_wmma<128, 64><<<cdiv((long long)waves * 32, B), B, 0, stream>>>(
            agg1, W1, h1, mtiles);
    }

    // ---- layer 2: D=64 -> 16 ----
    k_init_agg<64><<<cdiv((long long)nNodes * 64, B), B, 0, stream>>>(h1, deg, agg2, nNodes);
    k_edge_agg<64><<<cdiv(nEdges, 8), B, 0, stream>>>(h1, edges, deg, agg2, nEdges);
    {
        int mtiles = nNodes / 16;
        int waves = mtiles * (16 / 16);
        k_gemm_relu_wmma<64, 16><<<cdiv((long long)waves * 32, B), B, 0, stream>>>(
            agg2, W2, out, mtiles);
    }
}